// MambaBlock_69492570849825
// MI455X (gfx1250) — compile-verified
//
#include <hip/hip_runtime.h>
#include <hip/hip_bf16.h>
#include <math.h>

// ---------------------------------------------------------------------------
// Mamba block, fp32 end-to-end, WMMA f32 16x16x4 for all GEMMs (matches
// reference precision; op is scan-latency-bound so fp32 matrix path is free).
// Shapes: L=1024 d_model=128 d_inner=256 d_state=256 dt_rank=8 d_conv=4
// ---------------------------------------------------------------------------

typedef float v2f __attribute__((ext_vector_type(2)));
typedef float v8f __attribute__((ext_vector_type(8)));

__device__ __forceinline__ float silu_f(float x) { return x / (1.0f + __expf(-x)); }
__device__ __forceinline__ float softplus_f(float x) {
    return (x > 20.0f) ? x : log1pf(__expf(x));
}

// ---------------------------------------------------------------------------
// Generic one-wave-per-16x16-tile GEMM:  C[M,N] = A[M,K] @ W[N,K]^T (+bias)
// V_WMMA_F32_16X16X4_F32 operand layout (ISA 7.12.2, 32-bit A 16x4):
//   A: lanes 0-15 -> M=lane, VGPR0=K0,VGPR1=K1 ; lanes 16-31 -> K2,K3
//   B: mirrored (N across lanes, K split across halves)
//   C/D: VGPR r -> M=r (lanes 0-15) / M=r+8 (lanes 16-31), N=lane&15
// Ragged N handled branch-free: clamp W row for safe loads, zero via cndmask.
// mode: 0 = plain (+bias), 1 = softplus epilogue
// Grid must supply exactly (M/16)*ntilesN waves.
// ---------------------------------------------------------------------------
__global__ __launch_bounds__(256)
void k_gemm_wmma_f32(const float* __restrict__ A, int lda,
                     const float* __restrict__ W, int ldw,
                     float* __restrict__ C, int ldc,
                     int N, int K, int ntilesN,
                     const float* __restrict__ bias, int mode)
{
    const int lane = threadIdx.x & 31;
    const int wid  = (blockIdx.x * blockDim.x + threadIdx.x) >> 5;
    const int tileM = wid / ntilesN;
    const int tileN = wid % ntilesN;
    const int row0 = tileM * 16;
    const int col0 = tileN * 16;
    const int half = lane >> 4;   // 0: K pair {0,1}, 1: K pair {2,3}
    const int l16  = lane & 15;

    const int  colB   = col0 + l16;                  // column of W^T this lane feeds
    const bool colOK  = (colB < N);                  // ragged-N guard (N=520 case)
    const int  colIdx = colOK ? colB : (N - 1);      // clamped -> loads always legal

    const float* Arow = A + (row0 + l16) * lda;
    const float* Wrow = W + colIdx * ldw;

    v8f acc = {};
    #pragma unroll 4
    for (int k = 0; k < K; k += 4) {
        const int ka = k + half * 2;
        v2f a, b;
        a.x = Arow[ka + 0];
        a.y = Arow[ka + 1];
        const float bx = Wrow[ka + 0];               // unconditional b64 load
        const float by = Wrow[ka + 1];
        b.x = colOK ? bx : 0.0f;                     // v_cndmask, no branch
        b.y = colOK ? by : 0.0f;
        acc = __builtin_amdgcn_wmma_f32_16x16x4_f32(
                  /*neg_a=*/false, a, /*neg_b=*/false, b,
                  /*c_mod=*/(short)0, acc, /*reuse_a=*/false, /*reuse_b=*/false);
    }

    if (colOK) {
        const float bv = bias ? bias[colB] : 0.0f;
        #pragma unroll
        for (int r = 0; r < 8; ++r) {
            const int m = row0 + r + half * 8;
            float v = acc[r] + bv;
            if (mode == 1) v = softplus_f(v);
            C[m * ldc + colB] = v;
        }
    }
}

// ---------------------------------------------------------------------------
// Depthwise causal conv (k=4, left pad 3) + bias + SiLU.
// xr is the 1024x512 in_proj output; xs = columns [0,256).
// ---------------------------------------------------------------------------
__global__ __launch_bounds__(256)
void k_conv_silu(const float* __restrict__ xr,
                 const float* __restrict__ cw,   // 256 x 4
                 const float* __restrict__ cb,   // 256
                 float* __restrict__ xsc)        // 1024 x 256
{
    const int idx = blockIdx.x * blockDim.x + threadIdx.x;  // t*256 + d
    const int t = idx >> 8;
    const int d = idx & 255;
    float acc = cb[d];
    #pragma unroll
    for (int k = 0; k < 4; ++k) {
        const int tt = t - 3 + k;
        const float xv = (tt >= 0) ? xr[tt * 512 + d] : 0.0f;
        acc = fmaf(cw[d * 4 + k], xv, acc);
    }
    xsc[idx] = silu_f(acc);
}

// ---------------------------------------------------------------------------
// Selective scan. One wave per channel d (8 waves/WG, 32 WGs = 256 channels).
// Each lane keeps 8 states n = lane + 32*j in VGPRs for all 1024 steps.
// B[t,:]/C[t,:] (shared by every channel in the WG) staged into LDS via the
// CDNA5 async-to-LDS path (GLOBAL_LOAD_ASYNC_TO_LDS_B32, ASYNCcnt), with a
// speculative prefetch of the next row overlapping this step's VALU work.
// y2[t,d] = (sum_n h*C + u*D[d]) * silu(res[t,d])  written by lane 0.
// ---------------------------------------------------------------------------
__global__ __launch_bounds__(256)
void k_scan(const float* __restrict__ dbc,    // 1024 x 520 (B at +8, C at +264)
            const float* __restrict__ delta,  // 1024 x 256
            const float* __restrict__ xsc,    // u: 1024 x 256
            const float* __restrict__ xr,     // res: cols [256,512) of 1024x512
            const float* __restrict__ A_log,  // 256 x 256
            const float* __restrict__ Dp,     // 256
            float* __restrict__ y2)           // 1024 x 256
{
    __shared__ float sB[256];
    __shared__ float sC[256];

    const int tid  = threadIdx.x;
    const int lane = tid & 31;
    const int wave = tid >> 5;
    const int d    = blockIdx.x * 8 + wave;

    float h[8];
    float Aval[8];
    #pragma unroll
    for (int j = 0; j < 8; ++j) {
        h[j]    = 0.0f;
        Aval[j] = -__expf(A_log[d * 256 + lane + 32 * j]);
    }
    const float Dd = Dp[d];

    // LDS byte addresses for this thread's staging slots (addr[31:0] of the
    // generic pointer is the LDS byte offset on CDNA5).
    const unsigned ldsB = (unsigned)(size_t)&sB[tid];
    const unsigned ldsC = (unsigned)(size_t)&sC[tid];
    const unsigned offB = (8u   + (unsigned)tid) * 4u;   // byte offset in row
    const unsigned offC = (264u + (unsigned)tid) * 4u;

    for (int t = 0; t < 1024; ++t) {
        const float* row = dbc + t * 520;

        // Async-copy B/C rows to LDS (GVS mode: saddr=row, vaddr=byte offset,
        // vdst=LDS byte address). Tracked by ASYNCcnt.
        asm volatile("global_load_async_to_lds_b32 %0, %1, %2\n\t"
                     "global_load_async_to_lds_b32 %3, %4, %2"
                     :
                     : "v"(ldsB), "v"(offB), "s"(row), "v"(ldsC), "v"(offC)
                     : "memory");

        // Speculative prefetch of next timestep's row into GL2/L0.
        __builtin_prefetch(row + 520, 0, 0);

        asm volatile("s_wait_asynccnt 0x0" ::: "memory");
        __syncthreads();

        const float dt  = delta[t * 256 + d];
        const float u   = xsc[t * 256 + d];
        const float dtu = dt * u;

        float part = 0.0f;
        #pragma unroll
        for (int j = 0; j < 8; ++j) {
            const int n = lane + 32 * j;
            const float dA = __expf(dt * Aval[j]);
            h[j] = fmaf(dA, h[j], dtu * sB[n]);
            part = fmaf(h[j], sC[n], part);
        }
        // wave32 butterfly reduction
        #pragma unroll
        for (int off = 16; off > 0; off >>= 1)
            part += __shfl_xor(part, off, 32);

        if (lane == 0) {
            const float r = xr[t * 512 + 256 + d];
            y2[t * 256 + d] = (part + u * Dd) * silu_f(r);
        }
        __syncthreads();   // protect sB/sC before next timestep overwrites
    }
}

// ---------------------------------------------------------------------------
// Host-side orchestration (all on `stream`, scratch carved from d_ws).
// ---------------------------------------------------------------------------
extern "C" void kernel_launch(void* const* d_in, const int* in_sizes, int n_in,
                              void* d_out, int out_size, void* d_ws, size_t ws_size,
                              hipStream_t stream)
{
    const float* x      = (const float*)d_in[0];   // 1024 x 128
    const float* inW    = (const float*)d_in[1];   // 512 x 128
    const float* inb    = (const float*)d_in[2];   // 512
    const float* convW  = (const float*)d_in[3];   // 256 x 1 x 4
    const float* convb  = (const float*)d_in[4];   // 256
    const float* ssmW   = (const float*)d_in[5];   // 520 x 256
    const float* deltaW = (const float*)d_in[6];   // 256 x 8
    const float* A_log  = (const float*)d_in[7];   // 256 x 256
    const float* Dp     = (const float*)d_in[8];   // 256
    const float* outW   = (const float*)d_in[9];   // 128 x 256
    const float* outb   = (const float*)d_in[10];  // 128
    float* out = (float*)d_out;                    // 1024 x 128

    float* ws  = (float*)d_ws;
    float* XR  = ws;                    // 1024*512  in_proj output (xs|res)
    float* XSC = XR  + 1024 * 512;      // 1024*256  conv+silu output (u)
    float* DBC = XSC + 1024 * 256;      // 1024*520  ssm_in output (dt|B|C)
    float* DEL = DBC + 1024 * 520;      // 1024*256  softplus(delta)
    float* Y2  = DEL + 1024 * 256;      // 1024*256  scan output * silu(res)

    // 1) in_proj: XR = x @ inW^T + inb          (M=1024,N=512,K=128) 64x32 tiles
    k_gemm_wmma_f32<<<256, 256, 0, stream>>>(x, 128, inW, 128, XR, 512,
                                             512, 128, 32, inb, 0);
    // 2) depthwise conv + SiLU
    k_conv_silu<<<1024, 256, 0, stream>>>(XR, convW, convb, XSC);
    // 3) ssm_in: DBC = XSC @ ssmW^T             (M=1024,N=520,K=256) 64x33 tiles
    k_gemm_wmma_f32<<<264, 256, 0, stream>>>(XSC, 256, ssmW, 256, DBC, 520,
                                             520, 256, 33, nullptr, 0);
    // 4) delta: DEL = softplus(DBC[:, :8] @ deltaW^T)  (M=1024,N=256,K=8)
    k_gemm_wmma_f32<<<128, 256, 0, stream>>>(DBC, 520, deltaW, 8, DEL, 256,
                                             256, 8, 16, nullptr, 1);
    // 5) selective scan (+ u*D, * silu(res))
    k_scan<<<32, 256, 0, stream>>>(DBC, DEL, XSC, XR, A_log, Dp, Y2);
    // 6) out_proj: out = Y2 @ outW^T + outb     (M=1024,N=128,K=256)
    k_gemm_wmma_f32<<<64, 256, 0, stream>>>(Y2, 256, outW, 256, out, 128,
                                            128, 256, 8, outb, 0);
}